// AnnealedSampler_84963043049728
// MI455X (gfx1250) — compile-verified
//
#include <hip/hip_runtime.h>
#include <hip/hip_bf16.h>
#include <stdint.h>

#define VOCAB  50257
#define VPAD   50272          // 16 * 3142, padded vocab for tiling
#define DM     2048
#define BATCH  64
#define NSTEPS 100

typedef float v2f __attribute__((ext_vector_type(2)));
typedef float v8f __attribute__((ext_vector_type(8)));

// ---------------------------------------------------------------------------
// Kernel 1: logits[64][VPAD] = h[64][2048] @ W[50257][2048]^T + b
// One wave32 computes a 16x16 output tile via V_WMMA_F32_16X16X4_F32,
// looping K in steps of 4. Memory-bound on the single pass over W (412 MB).
// ---------------------------------------------------------------------------
__global__ __launch_bounds__(256) void logits_wmma_kernel(
    const float* __restrict__ hmat, const float* __restrict__ W,
    const float* __restrict__ bias, float* __restrict__ logits)
{
    const int wave = threadIdx.x >> 5;
    const int lane = threadIdx.x & 31;
    const int half = lane >> 4;     // 0: lanes 0-15, 1: lanes 16-31
    const int l    = lane & 15;

    const int nTiles = VPAD / 16;   // 3142
    const int tileId = blockIdx.x * 8 + wave;
    if (tileId >= 4 * nTiles) return;
    const int m0 = (tileId & 3) * 16;   // batch tile (4 tiles of 16 = 64)
    const int n0 = (tileId >> 2) * 16;  // vocab tile

    int nIdx = n0 + l;
    if (nIdx >= VOCAB) nIdx = VOCAB - 1;   // clamp loads; stores are guarded

    // A fragment: lane holds row m = m0+l, ks {2*half, 2*half+1} per step
    // B fragment: lane holds col n = nIdx,  ks {2*half, 2*half+1} per step
    const float* __restrict__ hrow = hmat + (size_t)(m0 + l) * DM + 2 * half;
    const float* __restrict__ wrow = W    + (size_t)nIdx     * DM + 2 * half;

    v8f c;
    const float bv = bias[nIdx];           // C starts as broadcast bias
    #pragma unroll
    for (int j = 0; j < 8; ++j) c[j] = bv;

    for (int k0 = 0; k0 < DM; k0 += 32) {
        __builtin_prefetch(wrow + k0 + 64, 0, 1);   // global_prefetch_b8 on W stream
        #pragma unroll
        for (int kk = 0; kk < 32; kk += 4) {
            v2f a = *(const v2f*)(hrow + k0 + kk);
            v2f b = *(const v2f*)(wrow + k0 + kk);
            // D = A(16x4,f32) x B(4x16,f32) + C(16x16,f32)
            c = __builtin_amdgcn_wmma_f32_16x16x4_f32(
                    /*neg_a=*/false, a, /*neg_b=*/false, b,
                    /*c_mod=*/(short)0, c, /*reuse_a=*/false, /*reuse_b=*/false);
        }
    }

    if (n0 + l < VOCAB) {
        #pragma unroll
        for (int j = 0; j < 8; ++j) {
            // D layout: VGPR j holds M = j + 8*half, N = l
            logits[(size_t)(m0 + j + 8 * half) * VPAD + (n0 + l)] = c[j];
        }
    }
}

// ---------------------------------------------------------------------------
// Counter-based RNG (deterministic, graph-safe)
// ---------------------------------------------------------------------------
__device__ __forceinline__ uint64_t splitmix64(uint64_t z) {
    z += 0x9E3779B97F4A7C15ULL;
    z = (z ^ (z >> 30)) * 0xBF58476D1CE4E5B9ULL;
    z = (z ^ (z >> 27)) * 0x94D049BB133111EBULL;
    return z ^ (z >> 31);
}
__device__ __forceinline__ float u01(uint32_t x) {
    return (float)(x >> 8) * 0x1p-24f + 0x1p-25f;  // uniform in (0,1)
}

// ---------------------------------------------------------------------------
// Kernel 2: per (step, row): argmax_v( logits + t*N(0,1) + Gumbel )
// One block per (step,row); logits row is L2-resident.
// ---------------------------------------------------------------------------
__global__ __launch_bounds__(256) void sample_kernel(
    const float* __restrict__ logits, int* __restrict__ samples)
{
    const int b = blockIdx.x & (BATCH - 1);
    const int s = blockIdx.x >> 6;
    const float t = 1.0f - 0.99f * (float)s / 99.0f;   // linspace(1.0, 0.01, 100)
    const float* __restrict__ row = logits + (size_t)b * VPAD;

    float best  = -INFINITY;
    int   bestV = VOCAB;
    for (int v = threadIdx.x; v < VOCAB; v += 256) {
        uint64_t key = ((uint64_t)(s * BATCH + b) << 32) | (uint32_t)v;
        uint64_t r0 = splitmix64(key);
        uint64_t r1 = splitmix64(key ^ 0xD1B54A32D192ED03ULL);
        float u1 = u01((uint32_t)r0);
        float u2 = u01((uint32_t)(r0 >> 32));
        float u3 = u01((uint32_t)r1);
        // Box-Muller normal * t  +  Gumbel for categorical sampling (TEMP=1)
        float nrm = sqrtf(-2.0f * __logf(u1)) * __cosf(6.28318530718f * u2);
        float gum = -__logf(-__logf(u3));
        float score = row[v] + nrm * t + gum;
        if (score > best || (score == best && v < bestV)) { best = score; bestV = v; }
    }

    __shared__ float sv[256];
    __shared__ int   si[256];
    sv[threadIdx.x] = best;
    si[threadIdx.x] = bestV;
    __syncthreads();
    for (int off = 128; off > 0; off >>= 1) {
        if (threadIdx.x < (unsigned)off) {
            float ov = sv[threadIdx.x + off];
            int   oi = si[threadIdx.x + off];
            if (ov > sv[threadIdx.x] || (ov == sv[threadIdx.x] && oi < si[threadIdx.x])) {
                sv[threadIdx.x] = ov;
                si[threadIdx.x] = oi;
            }
        }
        __syncthreads();
    }
    if (threadIdx.x == 0) samples[s * BATCH + b] = si[0];
}

// ---------------------------------------------------------------------------
// Kernel 3: mode over steps per row; smallest value on count ties.
// ---------------------------------------------------------------------------
__global__ void mode_kernel(const int* __restrict__ samples,
                            float* __restrict__ out)
{
    const int b = threadIdx.x;
    if (b >= BATCH) return;
    int bestVal = VOCAB;
    int bestCnt = 0;
    for (int i = 0; i < NSTEPS; ++i) {
        const int v = samples[i * BATCH + b];
        int c = 0;
        for (int j = 0; j < NSTEPS; ++j)
            c += (samples[j * BATCH + b] == v);
        if (c > bestCnt || (c == bestCnt && v < bestVal)) { bestCnt = c; bestVal = v; }
    }
    out[b] = (float)bestVal;   // < 2^24, exact in f32
}

// ---------------------------------------------------------------------------
extern "C" void kernel_launch(void* const* d_in, const int* in_sizes, int n_in,
                              void* d_out, int out_size, void* d_ws, size_t ws_size,
                              hipStream_t stream) {
    const float* hmat = (const float*)d_in[0];   // [64, 2048]
    const float* W    = (const float*)d_in[1];   // [50257, 2048]
    const float* bias = (const float*)d_in[2];   // [50257]

    float* logits  = (float*)d_ws;                              // 64*VPAD f32 (12.9 MB)
    int*   samples = (int*)((char*)d_ws +
                            (size_t)BATCH * VPAD * sizeof(float)); // 100*64 i32

    const int totalTiles = 4 * (VPAD / 16);          // 12568 wave-tiles
    const int blocks1    = (totalTiles + 7) / 8;     // 8 waves / 256-thread block

    logits_wmma_kernel<<<blocks1, 256, 0, stream>>>(hmat, W, bias, logits);
    sample_kernel<<<NSTEPS * BATCH, 256, 0, stream>>>(logits, samples);
    mode_kernel<<<1, 64, 0, stream>>>(samples, (float*)d_out);
}